// InstructionMoE_62380105007527
// MI455X (gfx1250) — compile-verified
//
#include <hip/hip_runtime.h>
#include <stdint.h>

typedef float v2f __attribute__((ext_vector_type(2)));
typedef float v4f __attribute__((ext_vector_type(4)));
typedef float v8f __attribute__((ext_vector_type(8)));

#define K_DIM          4096
#define BK             64                    // K-tile per pipeline stage
#define NT             (K_DIM / BK)          // 64 tiles
#define ROWS_PER_WAVE  16
#define WAVES_PER_WG   8
#define ROWS_PER_WG    (ROWS_PER_WAVE * WAVES_PER_WG)   // 128
#define A_STRIDE       68                    // 64+4 pad: b64-aligned, conflict-free A reads
#define A_TILE_FLOATS  (ROWS_PER_WAVE * A_STRIDE)       // 1088
#define W_STRIDE       4098                 // 4096+2 pad: de-conflicts the two W columns

// router_weights: [BATCH, K_DIM] f32   W_values: [K_DIM, 2] f32   out: [BATCH, 2] f32
__global__ __launch_bounds__(256)
void InstructionMoE_62380105007527_kernel(const float* __restrict__ A,
                                          const float* __restrict__ W,
                                          float* __restrict__ out)
{
    __shared__ float lds_a[2][WAVES_PER_WG][A_TILE_FLOATS]; // 69632 B (double-buffered)
    __shared__ float lds_wt[2 * W_STRIDE];                  // 32784 B, Wt[col][k] transposed

    const int tid  = threadIdx.x;
    const int wave = tid >> 5;
    const int lane = tid & 31;
    const int hlf  = lane >> 4;          // 0: lanes 0-15, 1: lanes 16-31
    const int l15  = lane & 15;
    const int rowBase = blockIdx.x * ROWS_PER_WG + wave * ROWS_PER_WAVE;

    // Per-lane addresses for the 8 async b128 copies of one 16xBK tile:
    // instruction i: lanes 0-15 -> row 2i (256B contiguous), lanes 16-31 -> row 2i+1.
    uint32_t goff[8], lbase[8];
    #pragma unroll
    for (int i = 0; i < 8; ++i) {
        const int row = 2 * i + hlf;
        goff[i]  = ((uint32_t)(rowBase + row) * K_DIM + l15 * 4) * 4u;
        lbase[i] = (uint32_t)(uintptr_t)&lds_a[0][wave][row * A_STRIDE + l15 * 4];
    }
    const uint32_t bufBytes =
        (uint32_t)((uintptr_t)&lds_a[1][0][0] - (uintptr_t)&lds_a[0][0][0]);

    // Prologue: issue tile 0 into buffer 0 (overlaps with the W preload below).
    #pragma unroll
    for (int i = 0; i < 8; ++i) {
        asm volatile("global_load_async_to_lds_b128 %0, %1, %2"
                     :: "v"(lbase[i]), "v"(goff[i]), "s"(A) : "memory");
    }

    // One-time W preload + transpose: Wt[c][k]. 8192 floats via coalesced b128 loads.
    #pragma unroll
    for (int i = 0; i < 8; ++i) {
        const int j = tid + 256 * i;          // float4 index: covers W rows 2j, 2j+1
        v4f f = ((const v4f*)W)[j];
        v2f c0 = { f.x, f.z };                // column 0 at k=2j, 2j+1
        v2f c1 = { f.y, f.w };                // column 1 at k=2j, 2j+1
        *(v2f*)&lds_wt[0 * W_STRIDE + 2 * j] = c0;
        *(v2f*)&lds_wt[1 * W_STRIDE + 2 * j] = c1;
    }
    __syncthreads();   // only barrier in the kernel: Wt becomes read-only

    v8f acc = {};

    for (int t = 0; t < NT; ++t) {
        if (t + 1 < NT) {
            // Issue tile t+1 into the other buffer, then wait for tile t only:
            // async loads retire in order, so ASYNCcnt <= 8 means tile t landed.
            const uint32_t lofs = (uint32_t)((t + 1) & 1) * bufBytes;
            const uint32_t gadd = (uint32_t)(t + 1) * (BK * 4u);
            #pragma unroll
            for (int i = 0; i < 8; ++i) {
                asm volatile("global_load_async_to_lds_b128 %0, %1, %2"
                             :: "v"(lbase[i] + lofs), "v"(goff[i] + gadd), "s"(A)
                             : "memory");
            }
            asm volatile("s_wait_asynccnt 8" ::: "memory");
        } else {
            asm volatile("s_wait_asynccnt 0" ::: "memory");
        }

        const float* at = &lds_a[t & 1][wave][0];
        const int k0 = t * BK;

        #pragma unroll
        for (int kk = 0; kk < BK / 4; ++kk) {
            const int kloc = kk * 4 + hlf * 2;

            // A operand (16x4 f32): VGPR0 = K kloc, VGPR1 = K kloc+1 (per half-wave).
            v2f a = *(const v2f*)&at[l15 * A_STRIDE + kloc];

            // B operand (4x16 f32): column N = l15; N=0,1 carry W, rest zero.
            // Unconditional clamped-address b64 load + cndmask (no EXEC divergence).
            v2f wv = *(const v2f*)&lds_wt[(l15 & 1) * W_STRIDE + k0 + kloc];
            v2f b = { (l15 < 2) ? wv.x : 0.0f,
                      (l15 < 2) ? wv.y : 0.0f };

            acc = __builtin_amdgcn_wmma_f32_16x16x4_f32(
                /*neg_a=*/false, a, /*neg_b=*/false, b,
                /*c_mod=*/(short)0, acc, /*reuse_a=*/false, /*reuse_b=*/false);
        }
    }

    // C layout: VGPR j holds M=j (lanes 0-15) / M=j+8 (lanes 16-31); N = l15.
    if (l15 < 2) {
        const int rbase = rowBase + hlf * 8;
        #pragma unroll
        for (int j = 0; j < 8; ++j) {
            out[(rbase + j) * 2 + l15] = acc[j];
        }
    }
}

extern "C" void kernel_launch(void* const* d_in, const int* in_sizes, int n_in,
                              void* d_out, int out_size, void* d_ws, size_t ws_size,
                              hipStream_t stream) {
    const float* A  = (const float*)d_in[0];   // router_weights [65536, 4096]
    const float* W  = (const float*)d_in[1];   // W_values [4096, 2]
    float* out      = (float*)d_out;           // [65536, 2]

    const int batch = in_sizes[0] / K_DIM;     // 65536
    dim3 grid(batch / ROWS_PER_WG);            // 512
    dim3 block(256);
    InstructionMoE_62380105007527_kernel<<<grid, block, 0, stream>>>(A, W, out);
}